// Sampler_18622978195933
// MI455X (gfx1250) — compile-verified
//
#include <hip/hip_runtime.h>
#include <stdint.h>
#include <math.h>

// Problem geometry (from reference): logits [64, 8, 128000] f32, argmax over
// vocab of the last sequence position -> 64 indices.
#define BATCH            64
#define SEQ              8
#define VOCAB            128000
#define CHUNKS_PER_ROW   25               // 25 blocks cooperate on one row
#define TILES_PER_CHUNK  5                // each block streams 5 LDS tiles
#define TILE_FLOATS      1024             // 256 threads * float4
#define CHUNK_FLOATS     (TILES_PER_CHUNK * TILE_FLOATS)   // 5120
#define TILE_BYTES       (TILE_FLOATS * 4)                 // 4096

// Monotone map: orders float32 (including sign/inf) as unsigned ints.
__device__ __forceinline__ uint32_t fmono(float f) {
    uint32_t b = __float_as_uint(f);
    return (b & 0x80000000u) ? ~b : (b | 0x80000000u);
}

// Pack (value, index) so unsigned-max == (max value, then SMALLEST index),
// matching jnp.argmax first-occurrence tie semantics. Done once per lane.
__device__ __forceinline__ unsigned long long pack_key(float v, uint32_t idx) {
    return ((unsigned long long)fmono(v) << 32) | (uint32_t)(~idx);
}

// CDNA5 async copy: per-lane 16B global -> LDS, tracked by ASYNCcnt.
// lds_addr: per-lane LDS byte address (low 32 bits of flat shared pointer),
// goff:     per-lane byte offset VGPR, base: uniform 64-bit SGPR base (GVS).
__device__ __forceinline__ void async_tile_load_b128(uint32_t lds_addr,
                                                     uint32_t goff,
                                                     const float* base) {
    asm volatile("global_load_async_to_lds_b128 %0, %1, %2"
                 :: "v"(lds_addr), "v"(goff), "s"(base)
                 : "memory");
}

__device__ __forceinline__ unsigned long long wave32_max(unsigned long long k) {
#pragma unroll
    for (int off = 16; off >= 1; off >>= 1) {
        unsigned long long o = __shfl_xor(k, off, 32);
        if (o > k) k = o;
    }
    return k;
}

__global__ void __launch_bounds__(256)
sampler_argmax_stage1(const float* __restrict__ logits,
                      unsigned long long* __restrict__ partials) {
    const int tid   = threadIdx.x;
    const int row   = blockIdx.x / CHUNKS_PER_ROW;
    const int chunk = blockIdx.x % CHUNKS_PER_ROW;

    // Last sequence position of this batch row, plus this block's chunk.
    const float* base = logits
        + (size_t)(row * SEQ + (SEQ - 1)) * (size_t)VOCAB
        + (size_t)chunk * CHUNK_FLOATS;

    __shared__ __align__(16) float tile[2][TILE_FLOATS];
    __shared__ unsigned long long wkeys[8];

    const uint32_t laneByte = (uint32_t)tid * 16u;
    const uint32_t ldsA0 = (uint32_t)(uintptr_t)&tile[0][tid * 4];
    const uint32_t ldsA1 = (uint32_t)(uintptr_t)&tile[1][tid * 4];

    // Prime the pipeline 2 deep: tile0 -> buf0, tile1 -> buf1.
    async_tile_load_b128(ldsA0, laneByte, base);
    async_tile_load_b128(ldsA1, (uint32_t)TILE_BYTES + laneByte, base);

    // Per-lane running argmax in plain f32. Lane scans strictly increasing
    // indices, so strict '>' preserves first-occurrence ties per lane.
    float    bv = -INFINITY;
    uint32_t bi = 0u;
    const uint32_t gbase = (uint32_t)(chunk * CHUNK_FLOATS + tid * 4);

#pragma unroll
    for (int t = 0; t < TILES_PER_CHUNK; ++t) {
        // Tile t has landed in LDS when <= (outstanding - this one) remain.
        if (t < TILES_PER_CHUNK - 1) {
            asm volatile("s_wait_asynccnt 0x1" ::: "memory");
        } else {
            asm volatile("s_wait_asynccnt 0x0" ::: "memory");
        }

        // Each lane consumes exactly the 16 bytes it async-loaded (no cross-
        // lane traffic, so no workgroup barrier needed) -> ds_load_b128.
        const float4 v = *(const float4*)&tile[t & 1][tid * 4];

        // Force the ds_load data into VGPRs so buffer (t&1) is provably dead
        // before the async engine may overwrite it with tile t+2 (async LDS
        // writes are unordered vs DScnt).
        asm volatile("s_wait_dscnt 0x0" ::: "memory");

        if (t + 2 < TILES_PER_CHUNK) {
            async_tile_load_b128((t & 1) ? ldsA1 : ldsA0,
                                 (uint32_t)((t + 2) * TILE_BYTES) + laneByte,
                                 base);
        }

        const uint32_t g = gbase + (uint32_t)(t * TILE_FLOATS);
        if (v.x > bv) { bv = v.x; bi = g + 0u; }
        if (v.y > bv) { bv = v.y; bi = g + 1u; }
        if (v.z > bv) { bv = v.z; bi = g + 2u; }
        if (v.w > bv) { bv = v.w; bi = g + 3u; }
    }

    // Pack once per lane; 64-bit key handles cross-lane/block tie-breaking.
    unsigned long long best = pack_key(bv, bi);

    // wave32 butterfly reduce, then 8-wave LDS reduce.
    best = wave32_max(best);
    if ((tid & 31) == 0) wkeys[tid >> 5] = best;
    __syncthreads();

    if (tid < 32) {
        unsigned long long b = (tid < 8) ? wkeys[tid] : 0ull;
#pragma unroll
        for (int off = 4; off >= 1; off >>= 1) {
            unsigned long long o = __shfl_xor(b, off, 32);
            if (o > b) b = o;
        }
        if (tid == 0) partials[blockIdx.x] = b;
    }
}

__global__ void __launch_bounds__(32)
sampler_argmax_stage2(const unsigned long long* __restrict__ partials,
                      float* __restrict__ out) {
    const int row  = blockIdx.x;
    const int lane = threadIdx.x;

    unsigned long long b =
        (lane < CHUNKS_PER_ROW) ? partials[row * CHUNKS_PER_ROW + lane] : 0ull;
    b = wave32_max(b);

    if (lane == 0) {
        const uint32_t idx = ~(uint32_t)b;      // recover vocab index
        out[row] = (float)idx;                  // harness output buffer is f32
    }
}

extern "C" void kernel_launch(void* const* d_in, const int* in_sizes, int n_in,
                              void* d_out, int out_size, void* d_ws, size_t ws_size,
                              hipStream_t stream) {
    (void)in_sizes; (void)n_in; (void)out_size; (void)ws_size;

    const float* logits = (const float*)d_in[0];
    unsigned long long* partials = (unsigned long long*)d_ws;  // 1600 * 8 B

    sampler_argmax_stage1<<<BATCH * CHUNKS_PER_ROW, 256, 0, stream>>>(logits, partials);
    sampler_argmax_stage2<<<BATCH, 32, 0, stream>>>(partials, (float*)d_out);
}